// GGNN_71124658422328
// MI455X (gfx1250) — compile-verified
//
#include <hip/hip_runtime.h>

typedef __attribute__((ext_vector_type(2))) float v2f;
typedef __attribute__((ext_vector_type(8))) float v8f;

#define D_FEAT 128   // D_IN == D_OUT == 128 in the reference

// ---------------------------------------------------------------------------
// Kernel 1: xw = x @ W.T  via V_WMMA_F32_16X16X4_F32 (fp32 exact).
// Block = 256 threads = 8 waves. Block covers 16 rows x 128 cols.
// Wave w computes the 16x16 tile at columns [16w, 16w+16).
// ---------------------------------------------------------------------------
__global__ __launch_bounds__(256) void ggnn_gemm_wmma(
    const float* __restrict__ x,   // [N,128]
    const float* __restrict__ W,   // [128,128] row-major: W[n,k]
    float* __restrict__ xw,        // [N,128]
    int nNodes) {
  const int lane = threadIdx.x & 31;
  const int wave = threadIdx.x >> 5;        // 0..7 -> column tile
  const int half = lane >> 4;               // 0 or 1 (K sub-pair select)
  const int l16  = lane & 15;
  const int row0 = blockIdx.x * 16;

  // A lane row (clamped for tail blocks; stores are predicated below)
  int mrow = row0 + l16;
  if (mrow >= nNodes) mrow = nNodes - 1;
  const int ncol = wave * 16 + l16;         // B column == output feature n

  const float* __restrict__ xrow = x + (size_t)mrow * D_FEAT;
  const float* __restrict__ wrow = W + (size_t)ncol * D_FEAT;   // W[n, :]

  v8f acc = {};
#pragma unroll
  for (int k = 0; k < D_FEAT; k += 4) {
    const int kk = k + half * 2;
    // A 16x4: VGPR0=K{0|2}, VGPR1=K{1|3} per half-wave -> contiguous pair
    v2f a = *(const v2f*)(xrow + kk);
    // B 4x16: B[k,n] = W[n,k] -> same contiguous K pair from W row n
    v2f b = *(const v2f*)(wrow + kk);
    // (neg_a, A, neg_b, B, c_mod, C, reuse_a, reuse_b)
    acc = __builtin_amdgcn_wmma_f32_16x16x4_f32(
        false, a, false, b, (short)0, acc, false, false);
  }

  // D 16x16 f32: VGPR v -> row (v + 8*half), col l16 of the tile
#pragma unroll
  for (int v = 0; v < 8; ++v) {
    const int r = row0 + half * 8 + v;
    if (r < nNodes) {
      xw[(size_t)r * D_FEAT + wave * 16 + l16] = acc[v];
    }
  }
}

// ---------------------------------------------------------------------------
// Kernel 2: per-edge gather + bias + scatter-add (native f32 global atomics).
// One wave per edge; each lane owns 4 contiguous features (float4 = 128).
// ---------------------------------------------------------------------------
__global__ __launch_bounds__(256) void ggnn_scatter_edges(
    const float* __restrict__ xw,    // [N,128]
    const int* __restrict__ src,     // [E]
    const int* __restrict__ dst,     // [E]
    const float* __restrict__ bias,  // [128]
    float* __restrict__ agg,         // [N,128], pre-zeroed
    int nEdges) {
  const int e = blockIdx.x * (blockDim.x >> 5) + (threadIdx.x >> 5);
  if (e >= nEdges) return;
  const int lane = threadIdx.x & 31;

  const int s = src[e];
  const int d = dst[e];

  const float4 xv = *(const float4*)(xw + (size_t)s * D_FEAT + lane * 4);
  const float4 bv = *(const float4*)(bias + lane * 4);
  float* dp = agg + (size_t)d * D_FEAT + lane * 4;

  unsafeAtomicAdd(dp + 0, xv.x + bv.x);
  unsafeAtomicAdd(dp + 1, xv.y + bv.y);
  unsafeAtomicAdd(dp + 2, xv.z + bv.z);
  unsafeAtomicAdd(dp + 3, xv.w + bv.w);
}

// ---------------------------------------------------------------------------
// Kernel 3: h = relu(agg); segment-sum over sorted `batch` into sums/counts.
// Each wave walks a contiguous node chunk, accumulating in registers and
// flushing with atomics only when the graph id changes (batch is sorted).
// ---------------------------------------------------------------------------
__global__ __launch_bounds__(256) void ggnn_relu_pool(
    const float* __restrict__ agg,    // [N,128]
    const int* __restrict__ batch,    // [N] sorted graph ids
    float* __restrict__ sums,         // [G,128], pre-zeroed
    float* __restrict__ counts,       // [G],     pre-zeroed
    int nNodes) {
  const int lane   = threadIdx.x & 31;
  const int waveId = blockIdx.x * (blockDim.x >> 5) + (threadIdx.x >> 5);
  const int nWaves = gridDim.x * (blockDim.x >> 5);
  const int chunk  = (nNodes + nWaves - 1) / nWaves;
  const int n0 = waveId * chunk;
  int n1 = n0 + chunk;
  if (n1 > nNodes) n1 = nNodes;
  if (n0 >= n1) return;

  float4 acc = make_float4(0.f, 0.f, 0.f, 0.f);
  int cnt = 0;
  int g = batch[n0];

  for (int n = n0; n < n1; ++n) {
    const int gn = batch[n];
    if (gn != g) {
      float* sp = sums + (size_t)g * D_FEAT + lane * 4;
      unsafeAtomicAdd(sp + 0, acc.x);
      unsafeAtomicAdd(sp + 1, acc.y);
      unsafeAtomicAdd(sp + 2, acc.z);
      unsafeAtomicAdd(sp + 3, acc.w);
      if (lane == 0) unsafeAtomicAdd(counts + g, (float)cnt);
      acc = make_float4(0.f, 0.f, 0.f, 0.f);
      cnt = 0;
      g = gn;
    }
    const float4 v = *(const float4*)(agg + (size_t)n * D_FEAT + lane * 4);
    acc.x += fmaxf(v.x, 0.f);
    acc.y += fmaxf(v.y, 0.f);
    acc.z += fmaxf(v.z, 0.f);
    acc.w += fmaxf(v.w, 0.f);
    ++cnt;
  }
  // final flush
  float* sp = sums + (size_t)g * D_FEAT + lane * 4;
  unsafeAtomicAdd(sp + 0, acc.x);
  unsafeAtomicAdd(sp + 1, acc.y);
  unsafeAtomicAdd(sp + 2, acc.z);
  unsafeAtomicAdd(sp + 3, acc.w);
  if (lane == 0) unsafeAtomicAdd(counts + g, (float)cnt);
}

// ---------------------------------------------------------------------------
// Kernel 4: out[g,f] = sums[g,f] / max(counts[g], 1)
// ---------------------------------------------------------------------------
__global__ __launch_bounds__(256) void ggnn_finalize(
    const float* __restrict__ sums,
    const float* __restrict__ counts,
    float* __restrict__ out, int total) {
  const int i = blockIdx.x * blockDim.x + threadIdx.x;
  if (i >= total) return;
  const float c = counts[i >> 7];  // i / 128
  out[i] = sums[i] / fmaxf(c, 1.0f);
}

// ---------------------------------------------------------------------------
// Host launcher
// ---------------------------------------------------------------------------
extern "C" void kernel_launch(void* const* d_in, const int* in_sizes, int n_in,
                              void* d_out, int out_size, void* d_ws, size_t ws_size,
                              hipStream_t stream) {
  const float* x          = (const float*)d_in[0];   // [N,128] fp32
  const int*   edge_index = (const int*)  d_in[1];   // [2,E] (int32 under jax default x64-off)
  const int*   batch      = (const int*)  d_in[2];   // [N] sorted
  const float* W          = (const float*)d_in[3];   // [128,128]
  const float* bias       = (const float*)d_in[4];   // [128]
  float*       out        = (float*)d_out;           // [G,128]

  const int N = in_sizes[0] / D_FEAT;   // 50000
  const int E = in_sizes[1] / 2;        // 500000
  const int G = out_size / D_FEAT;      // 64

  // Workspace layout (fp32): xw[N*128] | agg[N*128] | sums[G*128] | counts[G]
  float* xw     = (float*)d_ws;
  float* agg    = xw  + (size_t)N * D_FEAT;
  float* sums   = agg + (size_t)N * D_FEAT;
  float* counts = sums + (size_t)G * D_FEAT;

  // Zero the accumulation buffers every call (deterministic; capture-legal).
  hipMemsetAsync(agg, 0,
                 ((size_t)N * D_FEAT + (size_t)G * D_FEAT + (size_t)G) * sizeof(float),
                 stream);

  // 1) WMMA GEMM: xw = x @ W.T
  const int rowTiles = (N + 15) / 16;   // 3125 (exact, EXEC stays all-1s)
  ggnn_gemm_wmma<<<rowTiles, 256, 0, stream>>>(x, W, xw, N);

  // 2) Edge gather + bias + scatter-add (one wave per edge)
  const int wavesPerBlock = 256 / 32;
  ggnn_scatter_edges<<<(E + wavesPerBlock - 1) / wavesPerBlock, 256, 0, stream>>>(
      xw, edge_index, edge_index + E, bias, agg, E);

  // 3) ReLU + sorted-segment pooling (register accumulation, rare flushes)
  ggnn_relu_pool<<<128, 256, 0, stream>>>(agg, batch, sums, counts, N);

  // 4) Mean division
  const int total = G * D_FEAT;
  ggnn_finalize<<<(total + 255) / 256, 256, 0, stream>>>(sums, counts, out, total);
}